// GlobalAttentionPoolingPMG_43765716746325
// MI455X (gfx1250) — compile-verified
//
#include <hip/hip_runtime.h>
#include <math.h>

// MI455X (gfx1250) — wave32, WMMA (no MFMA).
// Plan: K0 init, K1 gate via v_wmma_f32_16x16x4_f32 + seg-max atomics,
//       K2 exp + seg-sum (reads only 4MB of gates), K3 fused alpha + readout.
// Memory-bound: ~1.03 GB total traffic -> ~45us at 23.3 TB/s.

typedef float v2f __attribute__((ext_vector_type(2)));
typedef float v8f __attribute__((ext_vector_type(8)));

__device__ __forceinline__ void atomicMaxF(float* addr, float val) {
    // int-ordering trick: float bits as signed int preserve order for >=0,
    // reversed (as unsigned) for <0. segmax initialized to -inf bits.
    int iv = __float_as_int(val);
    if (iv >= 0) atomicMax((int*)addr, iv);
    else         atomicMin((unsigned int*)addr, (unsigned int)iv);
}

// ---------------- K0: zero readout, init segmax/-inf and segsum/0 -------------
__global__ void k_init(float* __restrict__ readout, int readoutN,
                       float* __restrict__ segmax, float* __restrict__ segsum, int B) {
    int i = blockIdx.x * blockDim.x + threadIdx.x;
    if (i < readoutN) readout[i] = 0.0f;
    if (i < B) {
        ((int*)segmax)[i] = 0xFF800000;   // -inf
        segsum[i] = 0.0f;
    }
}

// ---------------- K1: gate = feat @ w + b via WMMA f32 16x16x4 ----------------
// One wave per 16-node tile. A = 16x4 feat slab, B = 4x16 with every column = w,
// so D[m][n] = gate[m] replicated across columns. 32 chained WMMAs (K=128).
__global__ void k_gate(const float* __restrict__ feat, const float* __restrict__ w,
                       const float* __restrict__ bgate, const int* __restrict__ seg,
                       float* __restrict__ gate, float* __restrict__ segmax,
                       int N, int D) {
    const int lane = threadIdx.x & 31;
    const int wave = threadIdx.x >> 5;
    const int tile = blockIdx.x * (blockDim.x >> 5) + wave;
    const int node0 = tile * 16;
    if (node0 >= N) return;                       // wave-uniform exit

    const float bb = bgate[0];

    if (node0 + 16 <= N) {                        // wave-uniform: EXEC all-1s for WMMA
        const int m = lane & 15;                  // matrix row this lane feeds
        const int halfSel = (lane >> 4) * 2;      // lanes 16-31 supply K = 4k+2, 4k+3
        const float* rowp = feat + (size_t)(node0 + m) * D + halfSel;
        const float* wp   = w + halfSel;          // B column n == w for all n
        v8f acc = {};
        const int steps = D >> 2;                 // K advances 4 per WMMA
        for (int k = 0; k < steps; ++k) {
            v2f a = *(const v2f*)(rowp + 4 * k);  // 8B aligned
            v2f b = *(const v2f*)(wp   + 4 * k);
            // (neg_a, A, neg_b, B, c_mod, C, reuse_a, reuse_b)
            acc = __builtin_amdgcn_wmma_f32_16x16x4_f32(
                      false, a, false, b, (short)0, acc, false, false);
        }
        // D layout: lane 0 column holds M=0..7 in acc[0..7]; lane 16 holds M=8..15.
        if (lane == 0 || lane == 16) {
            const int base = node0 + (lane >> 4) * 8;
            #pragma unroll
            for (int j = 0; j < 8; ++j) {
                float g = acc[j] + bb;
                gate[base + j] = g;
                atomicMaxF(&segmax[seg[base + j]], g);
            }
        }
    } else {
        // scalar tail (never taken for N % 16 == 0); lanes 0..15, one node each
        int n = node0 + lane;
        if (lane < 16 && n < N) {
            float g = bb;
            for (int d = 0; d < D; ++d) g += feat[(size_t)n * D + d] * w[d];
            gate[n] = g;
            atomicMaxF(&segmax[seg[n]], g);
        }
    }
}

// ---------------- K2: e = exp(gate - segmax), seg-sum ------------------------
__global__ void k_expsum(const float* __restrict__ gate, const float* __restrict__ segmax,
                         const int* __restrict__ seg, float* __restrict__ segsum, int N) {
    int n = blockIdx.x * blockDim.x + threadIdx.x;
    if (n >= N) return;
    int s = seg[n];
    float e = __expf(gate[n] - segmax[s]);
    atomicAdd(&segsum[s], e);
}

// ---------------- K3: alpha + weighted readout (sorted segments) -------------
// 128 threads/block = one thread per feature. Register accumulation; atomic
// flush only at segment boundaries (segment_ids sorted). Coalesced feat loads.
__global__ void k_readout(const float* __restrict__ feat, const float* __restrict__ gate,
                          const float* __restrict__ segmax, const float* __restrict__ segsum,
                          const int* __restrict__ seg, float* __restrict__ readout,
                          float* __restrict__ alpha, int N, int D) {
    const int d = threadIdx.x;                    // 0..D-1
    const long long per = ((long long)N + gridDim.x - 1) / gridDim.x;
    long long n0 = (long long)blockIdx.x * per;
    long long n1 = n0 + per; if (n1 > N) n1 = N;
    if (n0 >= n1) return;

    int curSeg = seg[n0];
    float acc = 0.0f;
    for (long long n = n0; n < n1; ++n) {
        int s = seg[n];                           // uniform per block iteration
        if (s != curSeg) {
            atomicAdd(&readout[(size_t)curSeg * D + d], acc);
            acc = 0.0f;
            curSeg = s;
        }
        float a = __expf(gate[n] - segmax[s]) / segsum[s];
        // stream-ahead hint -> global_prefetch_b8 (speculative; OOB is dropped)
        __builtin_prefetch(feat + (size_t)(n + 8) * D + d, 0, 3);
        acc += a * feat[(size_t)n * D + d];
        if (d == 0) alpha[n] = a;                 // alpha[:,None] flat
    }
    atomicAdd(&readout[(size_t)curSeg * D + d], acc);
}

extern "C" void kernel_launch(void* const* d_in, const int* in_sizes, int n_in,
                              void* d_out, int out_size, void* d_ws, size_t ws_size,
                              hipStream_t stream) {
    const float* feat = (const float*)d_in[0];   // [N, D]
    const float* w    = (const float*)d_in[1];   // [D, 1]
    const float* bg   = (const float*)d_in[2];   // [1]
    const int*   seg  = (const int*)d_in[3];     // [N] sorted

    const int D = in_sizes[1];
    const int N = in_sizes[3];
    const int B = (out_size - N) / D;

    float* readout = (float*)d_out;                      // [B, D]
    float* alpha   = (float*)d_out + (size_t)B * D;      // [N]

    float* gate   = (float*)d_ws;                        // N floats
    float* segmax = gate + N;                            // B floats
    float* segsum = segmax + B;                          // B floats

    // K0: init accumulators
    int initN = B * D > B ? B * D : B;
    k_init<<<(initN + 255) / 256, 256, 0, stream>>>(readout, B * D, segmax, segsum, B);

    // K1: WMMA gate + segment max. 8 waves (8 tiles) per 256-thread block.
    int tiles = (N + 15) / 16;
    int wavesPerBlock = 8;
    int blocks1 = (tiles + wavesPerBlock - 1) / wavesPerBlock;
    k_gate<<<blocks1, 32 * wavesPerBlock, 0, stream>>>(feat, w, bg, seg, gate, segmax, N, D);

    // K2: exp + segment sum (reads only gates, 4 MB)
    k_expsum<<<(N + 255) / 256, 256, 0, stream>>>(gate, segmax, seg, segsum, N);

    // K3: fused alpha + readout; 2048 blocks x D threads, ~489 nodes each
    k_readout<<<2048, D, 0, stream>>>(feat, gate, segmax, segsum, seg, readout, alpha, N, D);
}